// LSTM_16887811408430
// MI455X (gfx1250) — compile-verified
//
#include <hip/hip_runtime.h>
#include <cstdint>

typedef float v2f __attribute__((ext_vector_type(2)));
typedef float v8f __attribute__((ext_vector_type(8)));

#define WMMA_F32(a, b, c) \
    __builtin_amdgcn_wmma_f32_16x16x4_f32(false, (a), false, (b), (short)0, (c), false, false)

constexpr int Ts = 512, Bs = 64, Ds = 1024, Hs = 512, Ls = 3;
constexpr int G4H = 4 * Hs;  // 2048 gate columns (i,f,g,o)

// ---------------------------------------------------------------------------
// Kernel 1: XG = X * Wih^T + (bih + bhh), fp32 WMMA 16x16x4.
// M = T*B = 32768, N = 4H = 2048, K = D = 1024.
// dir==1 reads X time-flipped so XG rows are in the backward scan order.
// Workgroup: 128x128 output tile, 8 waves; wave = (row-pair group) x (col half),
// each wave owns 2x4 = 8 WMMA tiles (64 acc VGPRs).
// ---------------------------------------------------------------------------
__global__ __launch_bounds__(256, 1)
void xg_gemm(const float* __restrict__ X, const float* __restrict__ Wih,
             const float* __restrict__ bih, const float* __restrict__ bhh,
             float* __restrict__ XG, int dir) {
    const int tid   = threadIdx.x;
    const int lane  = tid & 31;
    const int wave  = tid >> 5;
    const int rp    = wave >> 1;      // 0..3 : 32-row group
    const int ch    = wave & 1;       // 0..1 : 64-col half
    const int n0    = blockIdx.x * 128;
    const int m0    = blockIdx.y * 128;
    const int l16   = lane & 15;
    const int khalf = lane >> 4;      // 0/1 -> K offset 0/2 (ISA A/B frag layout)

    // direction-aware source rows for the two A tiles of this wave
    int mA0 = m0 + rp * 32 + l16;
    int mA1 = mA0 + 16;
    auto srow = [&](int m) {
        int t = m >> 6, b = m & 63;
        int ts = dir ? (Ts - 1 - t) : t;
        return ts * Bs + b;
    };
    const float* pa0 = X + (size_t)srow(mA0) * Ds + 2 * khalf;
    const float* pa1 = X + (size_t)srow(mA1) * Ds + 2 * khalf;
    const float* pb[4];
#pragma unroll
    for (int c = 0; c < 4; ++c)
        pb[c] = Wih + (size_t)(n0 + ch * 64 + c * 16 + l16) * Ds + 2 * khalf;

    v8f acc0[4] = {v8f{}, v8f{}, v8f{}, v8f{}};
    v8f acc1[4] = {v8f{}, v8f{}, v8f{}, v8f{}};
    for (int kk = 0; kk < Ds; kk += 4) {
        v2f a0 = *(const v2f*)(pa0 + kk);
        v2f a1 = *(const v2f*)(pa1 + kk);
#pragma unroll
        for (int c = 0; c < 4; ++c) {
            v2f b   = *(const v2f*)(pb[c] + kk);
            acc0[c] = WMMA_F32(a0, b, acc0[c]);
            acc1[c] = WMMA_F32(a1, b, acc1[c]);
        }
    }
#pragma unroll
    for (int c = 0; c < 4; ++c) {
        int   col  = n0 + ch * 64 + c * 16 + l16;
        float badd = bih[col] + bhh[col];
#pragma unroll
        for (int v = 0; v < 8; ++v) {
            int row0 = m0 + rp * 32 + v + 8 * khalf;   // C/D layout: M = v (+8 hi lanes)
            XG[(size_t)row0 * G4H + col]        = acc0[c][v] + badd;
            XG[(size_t)(row0 + 16) * G4H + col] = acc1[c][v] + badd;
        }
    }
}

// ---------------------------------------------------------------------------
// Kernel 2: persistent recurrent LSTM for one (layer, direction).
// 16 WGs, each owns 32 h-columns (=128 gate cols). Per step: WMMA GEMM
// g += h_prev * Whh^T (M=64,N=128,K=512), gates staged in LDS, fused
// sigmoid/tanh + cell update, h written to ping-pong global buffer,
// device-scope atomic barrier across the 16 WGs.
// ---------------------------------------------------------------------------
__global__ __launch_bounds__(256, 1)
void lstm_rec(const float* __restrict__ XG, const float* __restrict__ Whh,
              const float* __restrict__ masks, float* __restrict__ Hbuf,
              float* __restrict__ o_allh, float* __restrict__ o_inp,
              float* __restrict__ o_hn, float* __restrict__ o_cn,
              unsigned* __restrict__ ctr, int layer, int dir) {
    constexpr int NWG = 16;
    const int wg    = blockIdx.x;
    const int hbase = wg * 32;            // owned h-columns [hbase, hbase+32)
    const int tid   = threadIdx.x;
    const int lane  = tid & 31;
    const int wave  = tid >> 5;
    const int r     = wave >> 1;          // tile-row 0..3 (16 batch rows each)
    const int ch    = wave & 1;           // col half: 4 gate-tile-cols each
    const int l16   = lane & 15;
    const int khalf = lane >> 4;

    __shared__ float gl[64][132];         // gate staging  (33 KB, padded)
    __shared__ float c_s[64][33];         // cell state    (8.25 KB, padded)
    for (int i = tid; i < 64 * 33; i += 256) (&c_s[0][0])[i] = 0.f;
    __syncthreads();

    const float* pb[4];
#pragma unroll
    for (int c = 0; c < 4; ++c) {
        int tc = ch * 4 + c;                                   // local tile-col 0..7
        int n  = (tc >> 1) * Hs + hbase + (tc & 1) * 16 + l16; // global gate row
        pb[c]  = Whh + (size_t)n * Hs + 2 * khalf;
    }
    const int arow = r * 16 + l16;

    for (int step = 0; step < Ts; ++step) {
        v8f acc[4] = {v8f{}, v8f{}, v8f{}, v8f{}};
        if (step > 0) {
            const float* rb = Hbuf + (size_t)((step + 1) & 1) * (Bs * Hs) +
                              (size_t)arow * Hs + 2 * khalf;
            for (int kk = 0; kk < Hs; kk += 4) {
                v2f a = *(const v2f*)(rb + kk);
#pragma unroll
                for (int c = 0; c < 4; ++c) {
                    v2f b  = *(const v2f*)(pb[c] + kk);
                    acc[c] = WMMA_F32(a, b, acc[c]);
                }
            }
        }
#pragma unroll
        for (int c = 0; c < 4; ++c) {
            int lcol = ch * 64 + c * 16 + l16;
#pragma unroll
            for (int v = 0; v < 8; ++v)
                gl[r * 16 + v + 8 * khalf][lcol] = acc[c][v];
        }
        __syncthreads();

        const int t_orig = dir ? (Ts - 1 - step) : step;
        float*    wbuf   = Hbuf + (size_t)(step & 1) * (Bs * Hs);
#pragma unroll
        for (int e = 0; e < 8; ++e) {
            int          id  = tid + e * 256;
            int          row = id >> 5, j = id & 31;
            const float* xg  = XG + ((size_t)step * Bs + row) * G4H + hbase + j;
            float iv = gl[row][j]      + xg[0];
            float fv = gl[row][32 + j] + xg[512];
            float gv = gl[row][64 + j] + xg[1024];
            float ov = gl[row][96 + j] + xg[1536];
            iv = 1.f / (1.f + __expf(-iv));
            fv = 1.f / (1.f + __expf(-fv));
            gv = tanhf(gv);
            ov = 1.f / (1.f + __expf(-ov));
            float m  = masks[(size_t)t_orig * Bs + row];
            float cn = (fv * c_s[row][j] + iv * gv) * m;   // blend with c0==0
            float hn = ov * tanhf(cn) * m;                 // blend with h0==0
            c_s[row][j]                = cn;
            wbuf[row * Hs + hbase + j] = hn;
            size_t ocol = (size_t)dir * Hs + hbase + j;
            o_allh[(((size_t)layer * Ts + t_orig) * Bs + row) * (2 * Hs) + ocol] = hn;
            if (layer == Ls - 1)
                o_inp[((size_t)t_orig * Bs + row) * (2 * Hs) + ocol] = hn;
            if (step == Ts - 1) {
                o_hn[((size_t)layer * Bs + row) * (2 * Hs) + ocol] = hn;
                o_cn[((size_t)layer * Bs + row) * (2 * Hs) + ocol] = cn;
            }
        }
        __threadfence();
        __syncthreads();
        if (tid == 0) {
            __hip_atomic_fetch_add(ctr, 1u, __ATOMIC_RELEASE,
                                   __HIP_MEMORY_SCOPE_AGENT);
            unsigned tgt = (unsigned)NWG * (unsigned)(step + 1);
            while (__hip_atomic_load(ctr, __ATOMIC_ACQUIRE,
                                     __HIP_MEMORY_SCOPE_AGENT) < tgt)
                __builtin_amdgcn_s_sleep(2);
        }
        __syncthreads();
    }
}

__global__ void init_ctrs(unsigned* c) {
    if (threadIdx.x < 8) c[threadIdx.x] = 0u;
}

// ---------------------------------------------------------------------------
extern "C" void kernel_launch(void* const* d_in, const int* in_sizes, int n_in,
                              void* d_out, int out_size, void* d_ws, size_t ws_size,
                              hipStream_t stream) {
    const float* x     = (const float*)d_in[0];
    const float* masks = (const float*)d_in[1];
    const float* w_ih  = (const float*)d_in[2];
    const float* w_hh  = (const float*)d_in[3];
    const float* b_ih  = (const float*)d_in[4];
    const float* b_hh  = (const float*)d_in[5];

    float*       out    = (float*)d_out;
    const size_t inp_sz = (size_t)Ts * Bs * 2 * Hs;   // [T,B,2H]
    const size_t hn_sz  = (size_t)Ls * Bs * 2 * Hs;   // [L,B,2H]
    float* o_inp  = out;
    float* o_hn   = out + inp_sz;
    float* o_cn   = o_hn + hn_sz;
    float* o_allh = o_cn + hn_sz;                     // [L,T,B,2H]

    float*       XG       = (float*)d_ws;             // [T,B,4H] fp32, reused
    const size_t xg_elems = (size_t)Ts * Bs * G4H;
    float*       Hbuf     = XG + xg_elems;            // 2 x [B,H] ping-pong
    unsigned*    ctrs     = (unsigned*)(Hbuf + 2 * (size_t)Bs * Hs);

    init_ctrs<<<1, 32, 0, stream>>>(ctrs);

    for (int layer = 0; layer < Ls; ++layer) {
        const float* Xin = (layer == 0) ? x : (o_allh + (size_t)(layer - 1) * inp_sz);
        for (int dir = 0; dir < 2; ++dir) {
            const float* wih = w_ih + ((size_t)layer * 2 + dir) * (size_t)G4H * Ds;
            const float* whh = w_hh + ((size_t)layer * 2 + dir) * (size_t)G4H * Hs;
            const float* bih = b_ih + ((size_t)layer * 2 + dir) * G4H;
            const float* bhh = b_hh + ((size_t)layer * 2 + dir) * G4H;
            xg_gemm<<<dim3(G4H / 128, (Ts * Bs) / 128), 256, 0, stream>>>(
                Xin, wih, bih, bhh, XG, dir);
            lstm_rec<<<dim3(16), 256, 0, stream>>>(
                XG, whh, masks, Hbuf, o_allh, o_inp, o_hn, o_cn,
                ctrs + (layer * 2 + dir), layer, dir);
        }
    }
}